// LAHLoss_40613210751079
// MI455X (gfx1250) — compile-verified
//
#include <hip/hip_runtime.h>
#include <hip/hip_bf16.h>
#include <stdint.h>

#define BB 2
#define CC 4
#define DD 128
#define HH 128
#define WW 128
#define TXD 16
#define TYD 16
#define ZCHUNK 32
#define NZCH (DD / ZCHUNK)        // 4
#define NTILES 64                 // 8x8 tiles of the 128x128 plane
#define NPART (NTILES * NZCH)     // 256 partial blocks per (b,c)
#define EPSF 1e-6f

typedef __attribute__((ext_vector_type(2))) float v2f;
typedef __attribute__((ext_vector_type(8))) float v8f;

// ---------- CDNA5 async global->LDS + split wait counters ----------
#if defined(__has_builtin)
#if __has_builtin(__builtin_amdgcn_s_wait_asynccnt)
#define WAIT_ASYNC_LE(n) __builtin_amdgcn_s_wait_asynccnt(n)
#endif
#if __has_builtin(__builtin_amdgcn_wmma_f32_16x16x4_f32)
#define HAVE_WMMA4 1
#endif
#endif
#ifndef WAIT_ASYNC_LE
#define WAIT_ASYNC_LE(n) asm volatile("s_wait_asynccnt %0" :: "i"(n) : "memory")
#endif

// GLOBAL_LOAD_ASYNC_TO_LDS_B32, GV mode: vdst = LDS byte offset, vaddr = 64-bit
// global address (low 32 bits of a generic __shared__ pointer are the
// LDS_BASE-relative byte offset, ISA 10.2 aperture layout).
__device__ __forceinline__ void async_g2l_b32(const void* g, void* l) {
  uint32_t loff  = (uint32_t)(uintptr_t)l;
  uint64_t gaddr = (uint64_t)(uintptr_t)g;
  asm volatile("global_load_async_to_lds_b32 %0, %1, off"
               :: "v"(loff), "v"(gaddr) : "memory");
}

__global__ __launch_bounds__(256) void lah_stencil_kernel(
    const float* __restrict__ x, const int* __restrict__ y,
    float* __restrict__ partials)
{
  __shared__ float    Xs[2][18][19];  // double-buffered raw-x halo slices
  __shared__ int      Ys[2][18][19];  // double-buffered class-id halo slices
  __shared__ float    pus[18][16];    // 3-pt x-products of u = 1-x
  __shared__ float    pgs[18][16];    // 3-pt x-AND of g = (y != c)
  __shared__ uint32_t trash[256];     // dummy async targets (uniform ASYNCcnt)
  __shared__ float    racc[256][4];   // per-thread partials for WMMA reduction
  __shared__ float    s16[16];

  const int tid = threadIdx.x;
  const int tx  = tid & 15;
  const int ty  = tid >> 4;

  const int tile = blockIdx.x & 63;
  const int zch  = blockIdx.x >> 6;
  const int bc   = blockIdx.y;            // b*4 + c
  const int b    = bc >> 2;
  const int c    = bc & 3;

  const int ty0 = (tile >> 3) * TYD;
  const int tx0 = (tile & 7) * TXD;
  const int z0  = zch * ZCHUNK;

  const size_t plane = (size_t)HH * WW;
  const float* xc = x + (size_t)bc * DD * plane;
  const int*   yb = y + (size_t)b  * DD * plane;

  const bool interior = (ty0 > 0) && (ty0 + TYD < HH) && (tx0 > 0) && (tx0 + TXD < WW);

  // ---- hoisted per-thread halo-fill addressing (324 = 256 + 68 elements) ----
  const int  yyA = tid / 18, xxA = tid - 18 * yyA;
  const int  gyA = ty0 + yyA - 1, gxA = tx0 + xxA - 1;
  const bool okA = (gyA >= 0) && (gyA < HH) && (gxA >= 0) && (gxA < WW);
  const size_t offA = (size_t)(okA ? gyA : 0) * WW + (size_t)(okA ? gxA : 0);

  const bool realB = (tid < 68);
  const int  eB  = 256 + tid;
  const int  yyB = eB / 18, xxB = eB - 18 * yyB;   // only meaningful when realB
  const int  gyB = ty0 + yyB - 1, gxB = tx0 + xxB - 1;
  const bool okB = realB && (gyB >= 0) && (gyB < HH) && (gxB >= 0) && (gxB < WW);
  const size_t offB = realB ? ((size_t)(okB ? gyB : 0) * WW + (size_t)(okB ? gxB : 0))
                            : offA;

  float* ldsXA[2] = { &Xs[0][yyA][xxA], &Xs[1][yyA][xxA] };
  int*   ldsYA[2] = { &Ys[0][yyA][xxA], &Ys[1][yyA][xxA] };
  float* ldsXB[2];
  int*   ldsYB[2];
  if (realB) {
    ldsXB[0] = &Xs[0][yyB][xxB]; ldsXB[1] = &Xs[1][yyB][xxB];
    ldsYB[0] = &Ys[0][yyB][xxB]; ldsYB[1] = &Ys[1][yyB][xxB];
  } else {
    ldsXB[0] = ldsXB[1] = (float*)&trash[tid];
    ldsYB[0] = ldsYB[1] = (int*)&trash[tid];
  }

  // exactly 4 async instructions per wave per slice in the interior path
  auto issue_fill = [&](int buf, int s) {
    const float* px_ = xc + (size_t)s * plane;
    const int*   py_ = yb + (size_t)s * plane;
    if (interior) {
      async_g2l_b32(px_ + offA, ldsXA[buf]);
      async_g2l_b32(py_ + offA, ldsYA[buf]);
      async_g2l_b32(px_ + offB, ldsXB[buf]);
      async_g2l_b32(py_ + offB, ldsYB[buf]);
    } else {
      const float xvA = px_[offA];        // offA clamped -> always safe
      const int   yvA = py_[offA];
      *ldsXA[buf] = okA ? xvA : 0.0f;     // OOB: x=0 -> u=1 (product identity)
      *ldsYA[buf] = okA ? yvA : -1;       // OOB: class -1 -> g=1 for every c
      if (realB) {
        const float xvB = px_[offB];
        const int   yvB = py_[offB];
        *ldsXB[buf] = okB ? xvB : 0.0f;
        *ldsYB[buf] = okB ? yvB : -1;
      }
      if (s + 1 < DD) __builtin_prefetch(px_ + plane + offA, 0, 3);
    }
  };

  // ---- software-pipelined z sweep with double-buffered async staging ----
  const int sFirst = (z0 > 0) ? (z0 - 1) : z0;
  const int sLast  = (z0 + ZCHUNK < DD) ? (z0 + ZCHUNK) : (DD - 1);

  float pm1u = 1.0f, pm1g = 1.0f, p0u = 1.0f, p0g = 1.0f;
  float uc = 0.0f, gc = 0.0f;
  float fp = 0.0f, fn = 0.0f, sx = 0.0f, sy = 0.0f;

  auto emit = [&](float p1u, float p1g) {   // output for slice holding (uc, gc)
    const float Eu = pm1u * p0u * p1u;      // 27-voxel product of u
    const float Eg = pm1g * p0g * p1g;      // 27-voxel AND of g
    const float xcen = 1.0f - uc;
    const float ycen = 1.0f - gc;
    fn += uc * (1.0f + Eu) * ycen;          // dm_hat * y_onehot
    fp += gc * (1.0f + Eg) * xcen;          // dm_gt  * x
    sx += xcen;
    sy += ycen;
  };

  int buf = 0;
  issue_fill(0, sFirst);                    // prologue: first slice in flight

  for (int s = sFirst; s <= sLast; ++s) {
    __syncthreads();                        // all readers of buf^1 & pus/pgs done
    const bool hasNext = (s < sLast);
    if (hasNext) issue_fill(buf ^ 1, s + 1);
    if (interior) {                         // wait: slice s complete, s+1 in flight
      if (hasNext) WAIT_ASYNC_LE(4);
      else         WAIT_ASYNC_LE(0);
    }
    __syncthreads();                        // slice-s LDS visible to all waves

    for (int e = tid; e < 288; e += 256) {  // 3-pt products along x
      const int r = e >> 4, cc2 = e & 15;
      const float a0 = Xs[buf][r][cc2], a1 = Xs[buf][r][cc2 + 1], a2 = Xs[buf][r][cc2 + 2];
      pus[r][cc2] = (1.0f - a0) * (1.0f - a1) * (1.0f - a2);
      const int b0 = Ys[buf][r][cc2], b1 = Ys[buf][r][cc2 + 1], b2 = Ys[buf][r][cc2 + 2];
      pgs[r][cc2] = (b0 != c && b1 != c && b2 != c) ? 1.0f : 0.0f;
    }
    __syncthreads();

    const float p1u = pus[ty][tx] * pus[ty + 1][tx] * pus[ty + 2][tx];
    const float p1g = pgs[ty][tx] * pgs[ty + 1][tx] * pgs[ty + 2][tx];
    const float ucn = 1.0f - Xs[buf][ty + 1][tx + 1];
    const float gcn = (Ys[buf][ty + 1][tx + 1] != c) ? 1.0f : 0.0f;

    if (s - 1 >= z0) emit(p1u, p1g);        // output z = s-1 (window complete)
    pm1u = p0u; p0u = p1u;
    pm1g = p0g; p0g = p1g;
    uc = ucn; gc = gcn;
    buf ^= 1;
  }
  if (z0 + ZCHUNK >= DD) emit(1.0f, 1.0f);  // z = DD-1 (slice DD is pad=1)

  // ---- block reduction: 256 threads x 4 quantities -> 4 sums ----
  __syncthreads();
  racc[tid][0] = fp; racc[tid][1] = fn; racc[tid][2] = sx; racc[tid][3] = sy;
  __syncthreads();

  const int pbase = (bc * NPART + (tile * NZCH + zch)) * 4;
#if defined(HAVE_WMMA4)
  // V_WMMA_F32_16X16X4_F32 with A = ones: D[m][n] = sum_k B[k][n] + C[m][n].
  // Quantity q owns columns 4q..4q+3 (n = lane&15); 16 chained WMMAs cover all
  // 256 threads once per quantity. fp32 accumulate, fixed order -> deterministic.
  if (tid < 32) {                           // wave 0 only, EXEC all ones
    const int lane = tid;
    const int q  = (lane & 15) >> 2;
    const int sI = lane & 3;
    const int h  = lane >> 4;
    const v2f ones = {1.0f, 1.0f};
    v8f acc = {};
#pragma unroll
    for (int i = 0; i < 16; ++i) {
      const int t0 = i * 16 + h * 8 + sI;
      v2f bv;
      bv[0] = racc[t0][q];
      bv[1] = racc[t0 + 4][q];
      acc = __builtin_amdgcn_wmma_f32_16x16x4_f32(false, ones, false, bv,
                                                  (short)0, acc, false, false);
    }
    if (lane < 16) s16[lane] = acc[0];      // D row 0: lane n holds column n
  }
  __syncthreads();
  if (tid == 0) {
    for (int q = 0; q < 4; ++q)
      partials[pbase + q] = s16[4 * q] + s16[4 * q + 1] + s16[4 * q + 2] + s16[4 * q + 3];
  }
#else
  for (int q = 0; q < 4; ++q) {
    __syncthreads();
    if (tid < 128) racc[tid][q] += racc[tid + 128][q];
    __syncthreads();
    for (int off = 64; off > 0; off >>= 1) {
      if (tid < off) racc[tid][q] += racc[tid + off][q];
      __syncthreads();
    }
    if (tid == 0) partials[pbase + q] = racc[0][q];
  }
#endif
}

__global__ void lah_finalize_kernel(const float* __restrict__ partials,
                                    float* __restrict__ out)
{
  __shared__ float sums[BB * CC][4];
  const int t = threadIdx.x;
  if (t < BB * CC * 4) {
    const int bc = t >> 2;
    const int q  = t & 3;
    float s = 0.0f;
    for (int p = 0; p < NPART; ++p)         // fixed order -> deterministic
      s += partials[(bc * NPART + p) * 4 + q];
    sums[bc][q] = s;
  }
  __syncthreads();
  if (t == 0) {
    float acc = 0.0f;
    for (int b = 0; b < BB; ++b)
      for (int c = 1; c < CC; ++c) {        // do_bg=False drops class 0
        const int bc = b * CC + c;
        const float fpn = sums[bc][0] / (sums[bc][2] + EPSF);
        const float fnn = sums[bc][1] / (sums[bc][3] + EPSF);
        acc += fmaxf(fpn, fnn);             // (a+b+|a-b|)/2 == max
      }
    out[0] = acc * (1.0f / 6.0f);
  }
}

extern "C" void kernel_launch(void* const* d_in, const int* in_sizes, int n_in,
                              void* d_out, int out_size, void* d_ws, size_t ws_size,
                              hipStream_t stream) {
  const float* x = (const float*)d_in[0];
  const int*   y = (const int*)d_in[1];
  float* out      = (float*)d_out;
  float* partials = (float*)d_ws;           // 8 * 256 * 4 floats = 32 KiB
  dim3 grid(NTILES * NZCH, BB * CC, 1);     // 2048 blocks x 256 threads
  lah_stencil_kernel<<<grid, 256, 0, stream>>>(x, y, partials);
  lah_finalize_kernel<<<1, 64, 0, stream>>>(partials, out);
  (void)in_sizes; (void)n_in; (void)out_size; (void)ws_size;
}